// LaplacianRegularization_32615981646503
// MI455X (gfx1250) — compile-verified
//
#include <hip/hip_runtime.h>
#include <hip/hip_bf16.h>
#include <stdint.h>

// ---------------------------------------------------------------------------
// Laplacian regularization, MI455X (gfx1250, wave32).
// Memory-bound: ~71 MB compulsory traffic -> ~3us @ 23.3 TB/s. Whole working
// set (~77 MB) is L2-resident (192 MB), so default RT cache policy everywhere.
// CDNA5 paths: global_load_async_to_lds_b128 (ASYNCcnt) double-buffered edge
// staging, s_wait_asynccnt, global_prefetch_b8, native f32/f64 global atomics.
// ---------------------------------------------------------------------------

#define WAVE_SIZE 32
#define BLOCK 256
#define WAVES_PER_BLOCK (BLOCK / WAVE_SIZE)
#define TILE 64  // edges per wave-stage: TILE*8B == 32 lanes * 16B == one b128 async op

#if defined(__HIP_DEVICE_COMPILE__) && __has_builtin(__builtin_amdgcn_global_load_async_to_lds_b128)
#define USE_ASYNC_LDS 1
#else
#define USE_ASYNC_LDS 0
#endif

// Builtin prototype (from hipcc diagnostic): param pointee is a 16-byte int
// vector; AS1 for the global source, AS3 for the LDS destination.
typedef int v4i32 __attribute__((vector_size(16)));
typedef __attribute__((address_space(1))) v4i32* global_b128_ptr;
typedef __attribute__((address_space(3))) v4i32* shared_b128_ptr;

__device__ __forceinline__ void wait_asynccnt_le2() {
#if defined(__HIP_DEVICE_COMPILE__)
#if __has_builtin(__builtin_amdgcn_s_wait_asynccnt)
  __builtin_amdgcn_s_wait_asynccnt(2);
#else
  asm volatile("s_wait_asynccnt 0x2" ::: "memory");
#endif
#endif
}

__device__ __forceinline__ void wait_asynccnt_0() {
#if defined(__HIP_DEVICE_COMPILE__)
#if __has_builtin(__builtin_amdgcn_s_wait_asynccnt)
  __builtin_amdgcn_s_wait_asynccnt(0);
#else
  asm volatile("s_wait_asynccnt 0x0" ::: "memory");
#endif
#endif
}

// Stage one TILE-edge chunk of rows+cols (int64) into this wave's LDS slice.
// lbuf points at [rows[TILE] | cols[TILE]] for the selected double-buffer half.
__device__ __forceinline__ void stage_tile(const long long* __restrict__ rows,
                                           const long long* __restrict__ cols,
                                           long long* lbuf, long long tile, int lane) {
  const long long* gr = rows + tile * TILE;
  const long long* gc = cols + tile * TILE;
  long long* lr = lbuf;
  long long* lc = lbuf + TILE;
#if USE_ASYNC_LDS
  // 32 lanes x 16B = 512B = TILE*8B per instruction; two async ops per tile.
  __builtin_amdgcn_global_load_async_to_lds_b128(
      (global_b128_ptr)((const char*)gr + lane * 16),
      (shared_b128_ptr)((char*)lr + lane * 16), 0, 0);
  __builtin_amdgcn_global_load_async_to_lds_b128(
      (global_b128_ptr)((const char*)gc + lane * 16),
      (shared_b128_ptr)((char*)lc + lane * 16), 0, 0);
#else
  int i0 = lane * 2;
  lr[i0]     = gr[i0];
  lr[i0 + 1] = gr[i0 + 1];
  lc[i0]     = gc[i0];
  lc[i0 + 1] = gc[i0 + 1];
#endif
}

__device__ __forceinline__ float edge_term(const float* __restrict__ y,
                                           const float* __restrict__ invd,
                                           int r, int c, float we) {
  float ir = invd[r];
  float ic = invd[c];
  const float4* yr = (const float4*)(y + (size_t)r * 16);  // 64B-aligned rows
  const float4* yc = (const float4*)(y + (size_t)c * 16);
  float s = 0.f;
#pragma unroll
  for (int j = 0; j < 4; ++j) {
    float4 a = yr[j];
    float4 b = yc[j];
    float d0 = a.x * ir - b.x * ic;
    float d1 = a.y * ir - b.y * ic;
    float d2 = a.z * ir - b.z * ic;
    float d3 = a.w * ir - b.w * ic;
    s = fmaf(d0, d0, s);
    s = fmaf(d1, d1, s);
    s = fmaf(d2, d2, s);
    s = fmaf(d3, d3, s);
  }
  return we * sqrtf(s);
}

__global__ void lap_init(float* __restrict__ deg, double* __restrict__ acc, int n) {
  int i = blockIdx.x * blockDim.x + threadIdx.x;
  if (i < n) deg[i] = 0.0f;
  if (i == 0) *acc = 0.0;
}

__global__ void lap_degree(const long long* __restrict__ rows,
                           const float* __restrict__ w,
                           float* __restrict__ deg, long long E) {
  long long stride = (long long)gridDim.x * blockDim.x;
  for (long long e = (long long)blockIdx.x * blockDim.x + threadIdx.x; e < E; e += stride) {
    int r = (int)rows[e];
    float we = w[e];
    // native global_atomic_add_f32, no return, agent scope
    __hip_atomic_fetch_add(&deg[r], we, __ATOMIC_RELAXED, __HIP_MEMORY_SCOPE_AGENT);
  }
}

__global__ void lap_invd(float* __restrict__ deg, int n) {
  int i = blockIdx.x * blockDim.x + threadIdx.x;
  if (i < n) deg[i] = 1.0f / sqrtf(deg[i]);
}

__global__ void __launch_bounds__(BLOCK)
lap_edges(const long long* __restrict__ rows, const long long* __restrict__ cols,
          const float* __restrict__ w, const float* __restrict__ y,
          const float* __restrict__ invd, double* __restrict__ acc, long long E) {
  // Per-wave slice: 2 double-buffer halves x (rows[TILE] + cols[TILE]) int64.
  __shared__ long long lds[WAVES_PER_BLOCK * 2 * 2 * TILE];  // 16 KB

  const int lane = threadIdx.x & (WAVE_SIZE - 1);
  const int wid = threadIdx.x >> 5;
  long long gw = (long long)blockIdx.x * WAVES_PER_BLOCK + wid;
  long long nwaves = (long long)gridDim.x * WAVES_PER_BLOCK;
  long long full_tiles = E / TILE;

  long long* wl = &lds[wid * (4 * TILE)];
  float lsum = 0.f;

  long long t0 = gw;
  if (t0 < full_tiles) stage_tile(rows, cols, wl, t0, lane);
  int cur = 0;
  for (long long t = t0; t < full_tiles; t += nwaves) {
    long long tn = t + nwaves;
    if (tn < full_tiles) {
      stage_tile(rows, cols, wl + (cur ^ 1) * (2 * TILE), tn, lane);
      __builtin_prefetch(&w[tn * TILE + lane * 2], 0, 3);  // global_prefetch_b8
      wait_asynccnt_le2();  // newest 2 ops outstanding -> current tile landed
    } else {
      wait_asynccnt_0();
    }
    const long long* tr = wl + cur * (2 * TILE);
    const long long* tc = tr + TILE;
    long long base = t * TILE;
#pragma unroll
    for (int k = 0; k < TILE / WAVE_SIZE; ++k) {
      int idx = lane + k * WAVE_SIZE;
      int r = (int)tr[idx];
      int c = (int)tc[idx];
      float we = w[base + idx];
      lsum += edge_term(y, invd, r, c, we);
    }
    cur ^= 1;
  }

  // Tail edges (E not a multiple of TILE): direct loads, bounds-checked.
  long long tail0 = full_tiles * TILE;
  for (long long e = tail0 + gw * WAVE_SIZE + lane; e < E; e += nwaves * WAVE_SIZE) {
    int r = (int)rows[e];
    int c = (int)cols[e];
    lsum += edge_term(y, invd, r, c, w[e]);
  }

  // wave32 reduce, then one f64 atomic per wave (block-local sums keep accuracy)
#pragma unroll
  for (int off = WAVE_SIZE / 2; off > 0; off >>= 1) lsum += __shfl_down(lsum, off, WAVE_SIZE);
  if (lane == 0)
    __hip_atomic_fetch_add(acc, (double)lsum, __ATOMIC_RELAXED, __HIP_MEMORY_SCOPE_AGENT);
}

__global__ void lap_final(const double* __restrict__ acc, float* __restrict__ out, long long E) {
  if (threadIdx.x == 0 && blockIdx.x == 0) *out = (float)(*acc / (double)E);
}

extern "C" void kernel_launch(void* const* d_in, const int* in_sizes, int n_in,
                              void* d_out, int out_size, void* d_ws, size_t ws_size,
                              hipStream_t stream) {
  const long long* edge_index = (const long long*)d_in[0];  // int64, shape (2, E)
  const float* w = (const float*)d_in[1];                   // (E,)
  const float* y = (const float*)d_in[2];                   // (N, 16)

  long long E = (long long)in_sizes[1];
  int N = in_sizes[2] / 16;
  const long long* rows = edge_index;
  const long long* cols = edge_index + E;

  float* deg = (float*)d_ws;                                    // N floats (becomes invd)
  size_t acc_off = ((size_t)N * sizeof(float) + 7) & ~(size_t)7;
  double* acc = (double*)((char*)d_ws + acc_off);               // 1 double
  float* out = (float*)d_out;

  int nb_nodes = (N + BLOCK - 1) / BLOCK;
  long long nb_deg_ll = (E + BLOCK - 1) / BLOCK;
  int nb_deg = nb_deg_ll > 4096 ? 4096 : (int)nb_deg_ll;

  lap_init<<<nb_nodes, BLOCK, 0, stream>>>(deg, acc, N);
  lap_degree<<<nb_deg, BLOCK, 0, stream>>>(rows, w, deg, E);
  lap_invd<<<nb_nodes, BLOCK, 0, stream>>>(deg, N);
  lap_edges<<<1024, BLOCK, 0, stream>>>(rows, cols, w, y, deg, acc, E);
  lap_final<<<1, 32, 0, stream>>>(acc, out, E);
}